// AdaptiveGraphConvolutionalLSTM_MW_53901839565709
// MI455X (gfx1250) — compile-verified
//
#include <hip/hip_runtime.h>
#include <math.h>

// ---------------- problem constants ----------------
#define BATCH   32
#define FDIM    1024
#define IN_DIM  7168                // 7*F
#define KSPLIT  4                   // K-split for WMMA GEMM
#define KCHUNK  (IN_DIM / KSPLIT)   // 1792, multiple of 4
#define NTILES  (FDIM / 16)         // 64 N-tiles per gate
#define FSPLIT  4                   // f-split for gc streaming
#define FCHUNK  (FDIM / FSPLIT)     // 256

typedef __attribute__((ext_vector_type(2))) float v2f;
typedef __attribute__((ext_vector_type(4))) float v4f;
typedef __attribute__((ext_vector_type(8))) float v8f;

// =====================================================================
// Kernel 1: fused graph-conv partials (R and NR share each mask read).
// Block = (batch b, mask k, f-chunk fs). 256 threads x 4 g-columns = full
// 1024-wide row, so every load (mask, wR, wNR) is a 16-byte b128 load.
// Mask stream (384MB total) is non-temporal; weights (24MB) L2-resident.
// Partials per f-chunk go to ws; reduce kernel sums them (deterministic).
// =====================================================================
__global__ __launch_bounds__(256)
void gc_partial_kernel(const float* __restrict__ x,
                       const float* __restrict__ m1, const float* __restrict__ m2,
                       const float* __restrict__ m3,
                       const float* __restrict__ wR1, const float* __restrict__ wR2,
                       const float* __restrict__ wR3,
                       const float* __restrict__ wNR1, const float* __restrict__ wNR2,
                       const float* __restrict__ wNR3,
                       float* __restrict__ pR, float* __restrict__ pNR)
{
    const int tid = threadIdx.x;
    int bid       = blockIdx.x;
    const int fs  = bid & (FSPLIT - 1);  bid >>= 2;
    const int k   = bid % 3;
    const int b   = bid / 3;
    const int g4  = tid * 4;             // first of 4 g columns
    const int f0  = fs * FCHUNK;

    __shared__ float xs[FCHUNK];
    xs[tid] = x[b * FDIM + f0 + tid];
    __syncthreads();

    const float* m   = (k == 0) ? m1  : (k == 1) ? m2  : m3;
    const float* wR  = (k == 0) ? wR1 : (k == 1) ? wR2 : wR3;
    const float* wNR = (k == 0) ? wNR1: (k == 1) ? wNR2: wNR3;

    const float* mp   = m   + (size_t)b * FDIM * FDIM + (size_t)f0 * FDIM + g4;
    const float* wRp  = wR  + (size_t)f0 * FDIM + g4;
    const float* wNRp = wNR + (size_t)f0 * FDIM + g4;

    v4f accR = {}, accNR = {};
    #pragma unroll 4
    for (int f = 0; f < FCHUNK; ++f) {
        v4f mv = __builtin_nontemporal_load((const v4f*)mp);  mp += FDIM;
        v4f wr = *(const v4f*)wRp;    wRp  += FDIM;
        v4f wn = *(const v4f*)wNRp;   wNRp += FDIM;
        v4f xm = xs[f] * mv;
        accR  += xm * wr;
        accNR += xm * wn;
    }

    const size_t o = ((size_t)(b * 3 + k) * FSPLIT + fs) * FDIM + g4;
    *(v4f*)(pR  + o) = accR;
    *(v4f*)(pNR + o) = accNR;
}

// =====================================================================
// Kernel 2: sum the FSPLIT partials and assemble combined[B, 7F]
// (gc_R | gc_NR | Hidden_State).
// =====================================================================
__global__ __launch_bounds__(256)
void gc_reduce_kernel(const float* __restrict__ pR, const float* __restrict__ pNR,
                      const float* __restrict__ hidden,
                      float* __restrict__ combined)
{
    const int idx = blockIdx.x * 256 + threadIdx.x;   // 0 .. B*IN_DIM
    if (idx >= BATCH * IN_DIM) return;
    const int b   = idx / IN_DIM;
    const int col = idx % IN_DIM;

    float v;
    if (col < 6 * FDIM) {
        const int nr = (col >= 3 * FDIM);
        const int k  = (col - nr * 3 * FDIM) / FDIM;
        const int g  = col & (FDIM - 1);
        const float* p = (nr ? pNR : pR) + ((size_t)(b * 3 + k) * FSPLIT) * FDIM + g;
        v = ((p[0] + p[FDIM]) + (p[2 * FDIM] + p[3 * FDIM]));
    } else {
        v = hidden[b * FDIM + (col - 6 * FDIM)];
    }
    combined[idx] = v;
}

// =====================================================================
// Kernel 3: fp32 WMMA GEMM  D[32, 4*1024] = combined[32,7168] @ W[g]^T
// One wave per (gate, n_tile, k_chunk). M=32 as two 16x16 accumulators
// sharing each B fragment; V_WMMA_F32_16X16X4_F32, K stepped by 4.
//
// Fragment layout (ISA 7.12.2, 32-bit A 16x4):
//   A: lane L<16 -> A[M=L, K={0,1}] in {v0,v1}; lane L+16 -> K={2,3}.
//   B: mirrored (lane n<16 -> B[K={0,1}, n]; lane n+16 -> K={2,3}).
//   C/D: lanes 0-15 -> M=v, N=lane;  lanes 16-31 -> M=v+8, N=lane-16.
// =====================================================================
__global__ __launch_bounds__(32)
void linear_wmma_kernel(const float* __restrict__ combined,
                        const float* __restrict__ W0, const float* __restrict__ W1,
                        const float* __restrict__ W2, const float* __restrict__ W3,
                        float* __restrict__ partial)
{
    const int lane = threadIdx.x;               // 0..31
    int bid  = blockIdx.x;
    const int ks   = bid % KSPLIT;  bid /= KSPLIT;
    const int nt   = bid % NTILES;  bid /= NTILES;
    const int gate = bid;                       // 0:f 1:i 2:o 3:C
    const float* W = (gate == 0) ? W0 : (gate == 1) ? W1 : (gate == 2) ? W2 : W3;

    const int row  = lane & 15;                 // M (A) / N (B) sub-index
    const int kk2  = (lane >> 4) * 2;           // K sub-offset {0,2}
    const int ncol = nt * 16 + row;

    v8f c0 = {}; v8f c1 = {};
    const int k0base = ks * KCHUNK;

    for (int k0 = k0base; k0 < k0base + KCHUNK; k0 += 4) {
        const float* ap0 = combined + (size_t)row        * IN_DIM + k0 + kk2;
        const float* ap1 = combined + (size_t)(row + 16) * IN_DIM + k0 + kk2;
        const float* bp  = W        + (size_t)ncol       * IN_DIM + k0 + kk2;  // B[k,n]=W[n,k]
        v2f a0, a1, bb;
        a0.x = ap0[0]; a0.y = ap0[1];
        a1.x = ap1[0]; a1.y = ap1[1];
        bb.x = bp[0];  bb.y = bp[1];
        c0 = __builtin_amdgcn_wmma_f32_16x16x4_f32(false, a0, false, bb, (short)0, c0, false, false);
        c1 = __builtin_amdgcn_wmma_f32_16x16x4_f32(false, a1, false, bb, (short)0, c1, false, false);
    }

    // Scatter D tiles into the partial buffer: [gate][ks][batch_row][col]
    const int n     = nt * 16 + (lane & 15);
    const int mbase = (lane < 16) ? 0 : 8;
    float* pb = partial + (size_t)(gate * KSPLIT + ks) * BATCH * FDIM;
    #pragma unroll
    for (int v = 0; v < 8; ++v) {
        pb[(size_t)(mbase + v)      * FDIM + n] = c0[v];   // batch rows 0..15
        pb[(size_t)(mbase + v + 16) * FDIM + n] = c1[v];   // batch rows 16..31
    }
}

// =====================================================================
// Kernel 4: reduce k-chunk partials, add bias, gates, LSTM update.
// out[0 : B*F) = Hidden, out[B*F : 2*B*F) = Cell.
// =====================================================================
__global__ __launch_bounds__(256)
void lstm_kernel(const float* __restrict__ partial,
                 const float* __restrict__ fl_b, const float* __restrict__ il_b,
                 const float* __restrict__ ol_b, const float* __restrict__ Cl_b,
                 const float* __restrict__ cell_in,
                 float* __restrict__ out)
{
    const int idx = blockIdx.x * 256 + threadIdx.x;     // b*F + g
    if (idx >= BATCH * FDIM) return;
    const int g = idx & (FDIM - 1);

    float s[4];
    #pragma unroll
    for (int gate = 0; gate < 4; ++gate) {
        float acc = 0.f;
        #pragma unroll
        for (int ks = 0; ks < KSPLIT; ++ks)
            acc += partial[(size_t)(gate * KSPLIT + ks) * BATCH * FDIM + idx];
        s[gate] = acc;
    }

    float f = 1.f / (1.f + __expf(-(s[0] + fl_b[g])));
    float i = 1.f / (1.f + __expf(-(s[1] + il_b[g])));
    float o = 1.f / (1.f + __expf(-(s[2] + ol_b[g])));
    float C = tanhf(s[3] + Cl_b[g]);

    float cell = f * cell_in[idx] + i * C;
    out[idx]                = o * tanhf(cell);
    out[BATCH * FDIM + idx] = cell;
}

// =====================================================================
extern "C" void kernel_launch(void* const* d_in, const int* in_sizes, int n_in,
                              void* d_out, int out_size, void* d_ws, size_t ws_size,
                              hipStream_t stream)
{
    const float* input  = (const float*)d_in[0];
    const float* R1_i   = (const float*)d_in[1];
    const float* R2_i   = (const float*)d_in[2];
    const float* R3_i   = (const float*)d_in[3];
    // d_in[4..6] = NR*_i masks: dead per reference NOTE (R masks reused)
    const float* Hidden = (const float*)d_in[7];
    const float* Cell   = (const float*)d_in[8];
    const float* wR1    = (const float*)d_in[9];
    const float* wR2    = (const float*)d_in[10];
    const float* wR3    = (const float*)d_in[11];
    const float* wNR1   = (const float*)d_in[12];
    const float* wNR2   = (const float*)d_in[13];
    const float* wNR3   = (const float*)d_in[14];
    const float* fl_w   = (const float*)d_in[15];
    const float* fl_b   = (const float*)d_in[16];
    const float* il_w   = (const float*)d_in[17];
    const float* il_b   = (const float*)d_in[18];
    const float* ol_w   = (const float*)d_in[19];
    const float* ol_b   = (const float*)d_in[20];
    const float* Cl_w   = (const float*)d_in[21];
    const float* Cl_b   = (const float*)d_in[22];
    float* out = (float*)d_out;

    // workspace layout (floats):
    //   combined : B*IN_DIM                  = 229376
    //   partial  : 4*KSPLIT*B*FDIM           = 524288
    //   pR       : B*3*FSPLIT*FDIM           = 393216
    //   pNR      : B*3*FSPLIT*FDIM           = 393216   (total ~6.2 MB)
    float* combined = (float*)d_ws;
    float* partial  = combined + (size_t)BATCH * IN_DIM;
    float* pR       = partial  + (size_t)4 * KSPLIT * BATCH * FDIM;
    float* pNR      = pR       + (size_t)BATCH * 3 * FSPLIT * FDIM;

    // 1) graph-conv streaming partials: B*3*FSPLIT = 384 blocks, b128 loads
    gc_partial_kernel<<<BATCH * 3 * FSPLIT, 256, 0, stream>>>(
        input, R1_i, R2_i, R3_i, wR1, wR2, wR3, wNR1, wNR2, wNR3, pR, pNR);

    // 2) assemble combined[B, 7F]
    gc_reduce_kernel<<<(BATCH * IN_DIM) / 256, 256, 0, stream>>>(
        pR, pNR, Hidden, combined);

    // 3) fp32 WMMA GEMM: 4 gates * 64 n-tiles * KSPLIT k-chunks, 1 wave each
    linear_wmma_kernel<<<4 * NTILES * KSPLIT, 32, 0, stream>>>(
        combined, fl_w, il_w, ol_w, Cl_w, partial);

    // 4) gate nonlinearity + LSTM state update
    lstm_kernel<<<(BATCH * FDIM) / 256, 256, 0, stream>>>(
        partial, fl_b, il_b, ol_b, Cl_b, Cell, out);
}